// BEVLiftNet_26929444946026
// MI455X (gfx1250) — compile-verified
//
#include <hip/hip_runtime.h>
#include <stdint.h>

// ---- problem constants (match reference) ----
#define Bb   2
#define Nn   4
#define Vv   (Bb*Nn)          // 8 camera views
#define Cc   64
#define Hh   128
#define Ww   352
#define HWp  (Hh*Ww)          // 45056
#define Xd   256
#define Yd   256
#define Zd   8
#define NVOX ((size_t)Bb*Zd*Xd*Yd)       // 1,048,576 voxels
#define ACC_FLOATS (NVOX*(size_t)Cc)     // 67,108,864 floats = 256 MB
#define NPTS (Vv*HWp)                     // 360,448 points

typedef float        v2f  __attribute__((ext_vector_type(2)));
typedef float        v8f  __attribute__((ext_vector_type(8)));
typedef unsigned int u32x4 __attribute__((ext_vector_type(4)));
typedef int          i32x4 __attribute__((ext_vector_type(4)));
typedef int          i32x8 __attribute__((ext_vector_type(8)));

// ---------------------------------------------------------------------------
// Kernel 1: zero the voxel accumulator (float4 grid-stride)
// ---------------------------------------------------------------------------
__global__ void k_zero(float4* __restrict__ p, size_t n4) {
    size_t i      = (size_t)blockIdx.x * blockDim.x + threadIdx.x;
    size_t stride = (size_t)gridDim.x * blockDim.x;
    float4 z = make_float4(0.f, 0.f, 0.f, 0.f);
    for (; i < n4; i += stride) p[i] = z;
}

// ---------------------------------------------------------------------------
// Kernel 2: WMMA projection.  For each point: ptsego = [Q|t] @ (d*w, d*h, d, 1)
// where Q = (T@E)[:3,:3] @ Kinv, t = (T@E)[:3,3].  One wave = 32 points via
// two V_WMMA_F32_16X16X4_F32 (A rows 0..2 = [Q|t], rest 0; B cols = points).
// D layout: lanes 0..15 hold rows 0..7 in VGPRs 0..7 -> x,y,z in D[0..2].
// ---------------------------------------------------------------------------
__global__ void k_geom(const float* __restrict__ depths,
                       const float* __restrict__ Kinv,   // (V,3,3), already K^-1
                       const float* __restrict__ E,      // (V,4,4)
                       const float* __restrict__ T,      // (V,4,4)
                       int* __restrict__ idx) {
    const int tid   = threadIdx.x;
    const int lane  = tid & 31;
    const int wid   = tid >> 5;
    const int wbase = (blockIdx.x * 8 + wid) * 32;   // HWp % 32 == 0 -> one view/wave
    const int v     = wbase / HWp;
    const int bcam  = v / Nn;

    // ---- A operand: 16x4, lane L holds A(L%16, k0) and A(L%16, k0+1),
    //      k0 = 0 for lanes 0..15, k0 = 2 for lanes 16..31 (ISA 7.12.2)
    const int r = lane & 15;
    float a0 = 0.f, a1 = 0.f;
    if (r < 3) {
        const float* Tv = T + v * 16;
        const float* Ev = E + v * 16;
        const float* Kv = Kinv + v * 9;
        float Mr[4];   // row r of (T @ E)
        #pragma unroll
        for (int j = 0; j < 4; ++j)
            Mr[j] = Tv[r*4+0]*Ev[0*4+j] + Tv[r*4+1]*Ev[1*4+j] +
                    Tv[r*4+2]*Ev[2*4+j] + Tv[r*4+3]*Ev[3*4+j];
        float Arow[4];
        Arow[0] = Mr[0]*Kv[0] + Mr[1]*Kv[3] + Mr[2]*Kv[6];
        Arow[1] = Mr[0]*Kv[1] + Mr[1]*Kv[4] + Mr[2]*Kv[7];
        Arow[2] = Mr[0]*Kv[2] + Mr[1]*Kv[5] + Mr[2]*Kv[8];
        Arow[3] = Mr[3];
        const int k0 = (lane < 16) ? 0 : 2;
        a0 = Arow[k0];
        a1 = Arow[k0 + 1];
    }
    v2f A; A.x = a0; A.y = a1;

    #pragma unroll
    for (int half = 0; half < 2; ++half) {
        const int n  = lane & 15;
        const int p  = wbase + half * 16 + n;
        const int hw = p - v * HWp;
        const float d  = depths[p];                 // depths flat == point index
        const float wx = (float)(hw % Ww);
        const float hy = (float)(hw / Ww);
        // B 4x16: VGPR0 = row K=0 (lanes 0-15) / K=2 (lanes 16-31), VGPR1 = K=1 / K=3
        v2f Bm;
        Bm.x = (lane < 16) ? d * wx : d;
        Bm.y = (lane < 16) ? d * hy : 1.0f;
        v8f Cz = {};
        v8f D = __builtin_amdgcn_wmma_f32_16x16x4_f32(
                    false, A, false, Bm, (short)0, Cz, false, false);
        if (lane < 16) {
            const float px = D[0], py = D[1], pz = D[2];
            // grid = int(mid + p/vox), vox=0.5 -> *2 ; (int) truncates like astype(int32)
            const int gx = (int)(128.0f + px * 2.0f);
            const int gy = (int)(128.0f + py * 2.0f);
            const int gz = (int)(4.0f   + pz * 2.0f);
            const bool ok = (gx >= 0) & (gx < Xd) & (gy >= 0) & (gy < Yd) &
                            (gz >= 0) & (gz < Zd);
            const int vox = ((bcam * Zd + gz) * Xd + gx) * Yd + gy;
            idx[p] = ok ? vox : -1;
        }
    }
}

// ---------------------------------------------------------------------------
// Kernel 3: scatter-add 64 channels per point.  feat reads are coalesced
// across lanes (consecutive hw); atomics hit 64 contiguous floats per point.
// ---------------------------------------------------------------------------
__global__ void k_scatter(const float* __restrict__ feat,
                          const int* __restrict__ idx,
                          float* __restrict__ acc) {
    const int p  = blockIdx.x * 256 + threadIdx.x;
    const int t  = idx[p];
    const int v  = p / HWp;
    const int hw = p - v * HWp;
    if (t >= 0) {
        const float* f = feat + (size_t)v * Cc * HWp + hw;
        float*       a = acc  + (size_t)t * Cc;
        #pragma unroll 8
        for (int c = 0; c < Cc; ++c)
            unsafeAtomicAdd(a + c, f[(size_t)c * HWp]);
    }
}

// ---------------------------------------------------------------------------
// Kernel 4: Z-max + flip + transpose.  Block = (b, x, 64-wide y tile).
// Each z-plane of the tile is one contiguous 16 KB row -> stream it into LDS
// with the Tensor Data Mover (double-buffered), max in registers, then
// transpose through padded LDS and store reversed-y coalesced.
// ---------------------------------------------------------------------------
__global__ __launch_bounds__(256) void k_reduce(const float* __restrict__ acc,
                                                float* __restrict__ out) {
    __shared__ float buf[2][4096];       // 2 x 16 KB ping-pong tiles
    __shared__ float stage[64 * 65];     // padded transpose staging

    const int tid = threadIdx.x;
    const int blk = blockIdx.x;
    const int b   = blk >> 10;           // / (256*4)
    const int rem = blk & 1023;
    const int x   = rem >> 2;
    const int y0  = (rem & 3) << 6;

    const uint64_t base_elem    = (((uint64_t)b * Zd * Xd + x) * Yd + y0) * Cc;
    const uint64_t zstride_elem = (uint64_t)Xd * Yd * Cc;

    // Build a 1-D TDM descriptor (D# groups per ISA §8.3/8.4) and issue the DMA.
    auto issue = [&](int z, int pb) {
        const uint64_t ga  = (uint64_t)(uintptr_t)acc +
                             (base_elem + (uint64_t)z * zstride_elem) * 4ull;
        const uint32_t lds = (uint32_t)(uintptr_t)(&buf[pb][0]); // low 32b = LDS offset
        u32x4 g0;
        g0[0] = 1u;                                        // count=1 (valid), load, no gather
        g0[1] = lds;                                       // lds_addr
        g0[2] = (uint32_t)(ga & 0xffffffffu);              // global_addr[31:0]
        g0[3] = (uint32_t)((ga >> 32) & 0x01ffffffu)       // global_addr[56:32]
              | (2u << 30);                                // type = 2 ("image")
        i32x8 g1;
        g1[0] = (int)(2u << 16);                           // data_size = 4 bytes
        g1[1] = (int)((4096u & 0xffffu) << 16);            // tensor_dim0[15:0]
        g1[2] = (int)((4096u >> 16) | (1u << 16));         // tensor_dim0[31:16] | tensor_dim1=1
        g1[3] = (int)(4096u << 16);                        // tile_dim0 = 4096
        g1[4] = 0;                                         // tile_dim1/2 unused
        g1[5] = 4096;                                      // tensor_dim0_stride
        g1[6] = 0;
        g1[7] = 0;
        const i32x4 zz4 = {0, 0, 0, 0};                    // groups 2/3 unused (<=2D)
        const i32x8 zz8 = {0, 0, 0, 0, 0, 0, 0, 0};        // extra group (6-arg form)
        __builtin_amdgcn_tensor_load_to_lds(g0, g1, zz4, zz4, zz8, 0);
    };

    float m[16];
    #pragma unroll
    for (int k = 0; k < 16; ++k) m[k] = -__builtin_huge_valf();

    if (tid < 32) issue(0, 0);                      // wave 0 drives the TDM
    for (int z = 0; z < Zd; ++z) {
        if (tid < 32) {
            if (z + 1 < Zd) {
                issue(z + 1, (z + 1) & 1);          // prefetch next plane
                __builtin_amdgcn_s_wait_tensorcnt((short)1);  // plane z done
            } else {
                __builtin_amdgcn_s_wait_tensorcnt((short)0);
            }
        }
        __syncthreads();
        const float* tzp = &buf[z & 1][0];
        #pragma unroll
        for (int k = 0; k < 16; ++k)
            m[k] = fmaxf(m[k], tzp[tid + 256 * k]); // conflict-free LDS reads
        __syncthreads();                            // before buffer reuse
    }

    // m[k] holds (y = (tid>>6) + 4k, c = tid&63).  Transpose via padded LDS.
    {
        const int c  = tid & 63;
        const int yq = tid >> 6;
        #pragma unroll
        for (int k = 0; k < 16; ++k)
            stage[c * 65 + yq + 4 * k] = m[k];
    }
    __syncthreads();

    // out[(b*C+c)*X + (255-x)][255-(y0+y)] ; lanes vary y -> coalesced stores
    const int y  = tid & 63;
    const int cq = tid >> 6;
    const int yo = 255 - (y0 + y);
    #pragma unroll
    for (int j = 0; j < 16; ++j) {
        const int c = cq * 16 + j;
        const size_t o = ((size_t)(b * Cc + c) * Xd + (255 - x)) * Yd + yo;
        out[o] = stage[c * 65 + y];
    }
}

// ---------------------------------------------------------------------------
extern "C" void kernel_launch(void* const* d_in, const int* in_sizes, int n_in,
                              void* d_out, int out_size, void* d_ws, size_t ws_size,
                              hipStream_t stream) {
    const float* feat   = (const float*)d_in[0];   // (8,64,128,352)
    const float* depths = (const float*)d_in[1];   // (2,4,128,352)
    const float* Kinv   = (const float*)d_in[2];   // (8,3,3)
    const float* E      = (const float*)d_in[3];   // (8,4,4)
    const float* T      = (const float*)d_in[4];   // (8,4,4)
    float* out = (float*)d_out;

    float* acc = (float*)d_ws;                                   // 256 MB accumulator
    int*   idx = (int*)((char*)d_ws + ACC_FLOATS * sizeof(float)); // 1.4 MB indices

    k_zero   <<<4096,        256, 0, stream>>>((float4*)acc, ACC_FLOATS / 4);
    k_geom   <<<NPTS / 256,  256, 0, stream>>>(depths, Kinv, E, T, idx);
    k_scatter<<<NPTS / 256,  256, 0, stream>>>(feat, idx, acc);
    k_reduce <<<Bb * Xd * (Yd / 64), 256, 0, stream>>>(acc, out);
    (void)in_sizes; (void)n_in; (void)out_size; (void)ws_size;
}